// RoPE2DEncoder_4234837754547
// MI455X (gfx1250) — compile-verified
//
#include <hip/hip_runtime.h>
#include <hip/hip_bf16.h>

// ---------------------------------------------------------------------------
// 2D RoPE cos/sin table generation for MI455X (gfx1250).
//
// Output: cos_2d [H*W, 128] then sin_2d [H*W, 128], fp32, concatenated in
// d_out. 256 MB of pure writes -> store-bandwidth bound (~11us @ 23.3 TB/s).
//
// Strategy:
//   Kernel A: build 4 tables (cosx/sinx/cosy/siny, each 512x32 fp32, 256 KB
//             total, L2-resident) using V_WMMA_F32_16X16X4_F32 to form the
//             pos (x) inv_freq outer-product phase tiles, then __sincosf.
//   Kernel B: stream the 256 MB output with coalesced nontemporal b128
//             stores; each lane does two 16B L2-hot table loads + 2 stores.
// ---------------------------------------------------------------------------

typedef float v2f __attribute__((ext_vector_type(2)));
typedef float v4f __attribute__((ext_vector_type(4)));
typedef float v8f __attribute__((ext_vector_type(8)));

#define RP_H     512
#define RP_W     512
#define RP_DIM   128
#define RP_HALF  32            // unique freqs per axis (dim_axis/2)
#define TBL_ELEMS 16384        // 512*32 floats per table
// ws float layout: [0)=cosx [16384)=sinx [32768)=cosy [49152)=siny

// ---------------------------------------------------------------------------
// Kernel A: phase tables via WMMA outer product.
// grid = 128 single-wave blocks; block t<64 -> x tables, t>=64 -> y tables.
// Tile tt: tileM = tt>>1 (16 pos rows), tileN = tt&1 (16 freq cols).
// A(16x4): pos in K=0 column only.  B(4x16): inv_freq in K=0 row only.
// D(16x16) = pos[M] * inv_freq[N].
// ---------------------------------------------------------------------------
__global__ void __launch_bounds__(32)
rope2d_tables_wmma(const float* __restrict__ invfx,
                   const float* __restrict__ invfy,
                   float* __restrict__ ws) {
    const int t     = blockIdx.x;        // 0..127
    const int lane  = threadIdx.x;       // 0..31 (wave32)
    const bool isY  = (t >= 64);
    const int  tt   = isY ? (t - 64) : t;
    const int tileM = tt >> 1;           // 0..31 -> pos base = tileM*16
    const int tileN = tt & 1;            // 0..1  -> col base = tileN*16
    const int l15   = lane & 15;
    const bool lo   = (lane < 16);

    const float* invf = isY ? invfy : invfx;
    // Unconditional in-bounds load (index <= 31), then lane-select -> EXEC
    // stays all-ones across the WMMA (required by ISA).
    const float fv = invf[tileN * 16 + l15];

    v2f a, b;
    v8f c = {};
    // A 16x4 f32: VGPR0 = K0 (lanes 0-15) / K2 (lanes 16-31); VGPR1 = K1/K3.
    a.x = lo ? (float)(tileM * 16 + l15) : 0.0f;   // K=0 column = pos
    a.y = 0.0f;
    // B 4x16 f32: VGPR0 = K0 row (lanes 0-15) / K2 row (lanes 16-31).
    b.x = lo ? fv : 0.0f;                          // K=0 row = inv_freq
    b.y = 0.0f;

    v8f d = __builtin_amdgcn_wmma_f32_16x16x4_f32(
        /*neg_a=*/false, a, /*neg_b=*/false, b,
        /*c_mod=*/(short)0, c, /*reuse_a=*/false, /*reuse_b=*/false);

    float* cosT = ws + (isY ? 2 * TBL_ELEMS : 0);
    float* sinT = cosT + TBL_ELEMS;
    const int col   = tileN * 16 + l15;
    const int rbase = tileM * 16 + (lo ? 0 : 8);   // D: VGPR v -> M = v (+8 hi lanes)

#pragma unroll
    for (int v = 0; v < 8; ++v) {
        float s, cc;
        __sincosf(d[v], &s, &cc);
        const int idx = (rbase + v) * RP_HALF + col;
        cosT[idx] = cc;
        sinT[idx] = s;
    }
}

// ---------------------------------------------------------------------------
// Kernel B: stream 256 MB output. One wave per output row (128 ch cos+sin).
// Lane L covers channels [4L, 4L+4): ch<64 -> x-table[w], ch>=64 -> y-table[h]
// (channel duplication folds via &31). Two coalesced NT b128 stores per lane.
// ---------------------------------------------------------------------------
__global__ void __launch_bounds__(256)
rope2d_stream(const float* __restrict__ ws, float* __restrict__ out) {
    const int lane = threadIdx.x & 31;
    const int wave = threadIdx.x >> 5;
    const int row  = blockIdx.x * 8 + wave;        // 0..H*W-1, row = h*W + w
    const int h    = row >> 9;                     // W = 512
    const int w    = row & (RP_W - 1);
    const int c0   = lane << 2;                    // channel base, mult of 4
    const bool isY = (c0 >= 64);
    const int pos  = isY ? h : w;

    const float* cosT = ws + (isY ? 2 * TBL_ELEMS : 0) + pos * RP_HALF + (c0 & 31);
    const float* sinT = cosT + TBL_ELEMS;
    const v4f cv = *reinterpret_cast<const v4f*>(cosT);   // 16B aligned, L2-hot
    const v4f sv = *reinterpret_cast<const v4f*>(sinT);

    float* outc = out + (size_t)row * RP_DIM + c0;
    float* outs = outc + (size_t)RP_H * RP_W * RP_DIM;
    __builtin_nontemporal_store(cv, reinterpret_cast<v4f*>(outc));
    __builtin_nontemporal_store(sv, reinterpret_cast<v4f*>(outs));
}

// ---------------------------------------------------------------------------
// Fallback: direct per-element sincos (used only if d_ws is too small).
// ---------------------------------------------------------------------------
__global__ void __launch_bounds__(256)
rope2d_direct(const float* __restrict__ invfx,
              const float* __restrict__ invfy,
              float* __restrict__ out) {
    const int lane = threadIdx.x & 31;
    const int wave = threadIdx.x >> 5;
    const int row  = blockIdx.x * 8 + wave;
    const int h    = row >> 9;
    const int w    = row & (RP_W - 1);
    const int c0   = lane << 2;
    const bool isY = (c0 >= 64);
    const float p  = (float)(isY ? h : w);
    const float* invf = isY ? invfy : invfx;

    v4f cv, sv;
#pragma unroll
    for (int j = 0; j < 4; ++j) {
        float s, cc;
        __sincosf(p * invf[(c0 & 31) + j], &s, &cc);
        cv[j] = cc;
        sv[j] = s;
    }
    float* outc = out + (size_t)row * RP_DIM + c0;
    float* outs = outc + (size_t)RP_H * RP_W * RP_DIM;
    __builtin_nontemporal_store(cv, reinterpret_cast<v4f*>(outc));
    __builtin_nontemporal_store(sv, reinterpret_cast<v4f*>(outs));
}

extern "C" void kernel_launch(void* const* d_in, const int* in_sizes, int n_in,
                              void* d_out, int out_size, void* d_ws, size_t ws_size,
                              hipStream_t stream) {
    // inputs: [0]=x (unused values), [1]=inv_freq_x[32], [2]=inv_freq_y[32],
    //         [3]=height, [4]=width  (fixed 512x512, DIM=128 per reference)
    const float* invfx = (const float*)d_in[1];
    const float* invfy = (const float*)d_in[2];
    float* out = (float*)d_out;

    const size_t needed = (size_t)4 * TBL_ELEMS * sizeof(float); // 256 KB
    const int nblocks = (RP_H * RP_W) / 8;                       // 32768

    if (ws_size >= needed) {
        rope2d_tables_wmma<<<128, 32, 0, stream>>>(invfx, invfy, (float*)d_ws);
        rope2d_stream<<<nblocks, 256, 0, stream>>>((const float*)d_ws, out);
    } else {
        rope2d_direct<<<nblocks, 256, 0, stream>>>(invfx, invfy, out);
    }
}